// RoiPooling_20684562497638
// MI455X (gfx1250) — compile-verified
//
#include <hip/hip_runtime.h>
#include <math.h>

// Problem constants from the reference
#define FDIM 14
#define CDIM 512
#define HDIM 7
#define WDIM 7
#define MAXW 16        // padded column capacity (rw <= 13 always since coords in [0,13])
#define ROWS_CHUNK 2   // rows staged per LDS chunk: 2*16*512*4B = 64 KB LDS

typedef float float4v __attribute__((ext_vector_type(4)));
typedef int   int4v   __attribute__((ext_vector_type(4)));
typedef __attribute__((address_space(1))) int4v* global_i4_ptr;
typedef __attribute__((address_space(3))) int4v* lds_i4_ptr;

#if defined(__has_builtin)
#if __has_builtin(__builtin_amdgcn_global_load_async_to_lds_b128) && \
    __has_builtin(__builtin_amdgcn_s_wait_asynccnt)
#define USE_ASYNC_LDS 1
#else
#define USE_ASYNC_LDS 0
#endif
#else
#define USE_ASYNC_LDS 0
#endif

__device__ __forceinline__ int imin(int a, int b) { return a < b ? a : b; }

__device__ __forceinline__ float4v max4(float4v a, float4v b) {
  float4v r;
  r.x = fmaxf(a.x, b.x);
  r.y = fmaxf(a.y, b.y);
  r.z = fmaxf(a.z, b.z);
  r.w = fmaxf(a.w, b.w);
  return r;
}

// One block per (roi n, bin-row bh). 128 threads; thread t owns channels [4t, 4t+4).
__global__ __launch_bounds__(128)
void roi_pool_kernel(const float* __restrict__ fm, const float* __restrict__ rois,
                     float* __restrict__ out) {
  __shared__ float buf[ROWS_CHUNK * MAXW * CDIM];  // 64 KB staging buffer

  const int tid = threadIdx.x;        // channel float4 index, 0..127
  const int bid = blockIdx.x;
  const int n   = bid / HDIM;
  const int bh  = bid % HDIM;

  // --- uniform ROI decode (TF-faithful binning) ---
  const float* rp = rois + 4 * n;
  const int x_min = (int)floorf((float)FDIM * rp[0]);
  const int y_min = (int)floorf((float)FDIM * rp[1]);
  const int x_max = (int)floorf((float)FDIM * rp[2]);
  const int y_max = (int)floorf((float)FDIM * rp[3]);
  const int rw = x_max - x_min;
  const int rh = y_max - y_min;
  const int sw = (rw > 0) ? (rw / WDIM) : 0;
  const int sh = (rh > 0) ? (rh / HDIM) : 0;

  // Row range belonging to bin-row bh.
  // sh>0: rows [bh*sh, (bh+1)*sh) (+ tail rows for bh==H-1, via min(ry//sh, H-1))
  // sh==0: all rows map to bin H-1; other bins empty.
  int ryS = bh * sh;
  int ryE = (bh == HDIM - 1) ? rh : imin((bh + 1) * sh, rh);
  if (ryE < ryS) ryE = ryS;

  // Accumulators: one float4 per column bin, init -inf (empty bins stay -inf).
  float4v acc[WDIM];
#pragma unroll
  for (int bw = 0; bw < WDIM; ++bw) {
    acc[bw].x = -INFINITY; acc[bw].y = -INFINITY;
    acc[bw].z = -INFINITY; acc[bw].w = -INFINITY;
  }

  const float4v NEG_INF = acc[0];

  for (int base = ryS; base < ryE; base += ROWS_CHUNK) {
    const int nr = imin(ROWS_CHUNK, ryE - base);

    // --- stage nr rows x rw cols of the feature map into LDS (async DMA path) ---
    for (int cr = 0; cr < nr; ++cr) {
      const int y = y_min + base + cr;
      const bool yok = (y >= 0) && (y < FDIM);
      for (int rx = 0; rx < rw; ++rx) {
        const int x = x_min + rx;
        float* lp = &buf[(cr * MAXW + rx) * CDIM + tid * 4];
        if (yok && x >= 0 && x < FDIM) {
          const float* gp = fm + ((y * FDIM + x) * CDIM + tid * 4);
#if USE_ASYNC_LDS
          __builtin_amdgcn_global_load_async_to_lds_b128(
              (global_i4_ptr)gp, (lds_i4_ptr)lp,
              /*imm offset=*/0, /*cpol=*/0);
#else
          *(float4v*)lp = *(const float4v*)gp;
#endif
        } else {
          // pixel outside the feature map: contributes nothing
          *(float4v*)lp = NEG_INF;
        }
      }
    }
#if USE_ASYNC_LDS
    __builtin_amdgcn_s_wait_asynccnt(0);
#endif
    __syncthreads();

    // --- max-reduce the staged strip into the 7 column-bin accumulators ---
#pragma unroll
    for (int bw = 0; bw < WDIM; ++bw) {
      const int cS = bw * sw;
      int cE = (bw == WDIM - 1) ? rw : imin((bw + 1) * sw, rw);
      if (cE < cS) cE = cS;
      for (int rx = cS; rx < cE; ++rx) {
        for (int cr = 0; cr < nr; ++cr) {
          const float4v v = *(const float4v*)&buf[(cr * MAXW + rx) * CDIM + tid * 4];
          acc[bw] = max4(acc[bw], v);
        }
      }
    }
    __syncthreads();  // protect LDS before next chunk overwrites it
  }

  // --- streaming non-temporal stores of the 7 bins (output never re-read) ---
  float* op = out + ((size_t)(n * HDIM + bh) * WDIM) * CDIM + tid * 4;
#pragma unroll
  for (int bw = 0; bw < WDIM; ++bw) {
    __builtin_nontemporal_store(acc[bw], (float4v*)(op + bw * CDIM));
  }
}

extern "C" void kernel_launch(void* const* d_in, const int* in_sizes, int n_in,
                              void* d_out, int out_size, void* d_ws, size_t ws_size,
                              hipStream_t stream) {
  (void)n_in; (void)out_size; (void)d_ws; (void)ws_size;
  const float* fm   = (const float*)d_in[0];   // [1,14,14,512] fp32
  const float* rois = (const float*)d_in[1];   // [1,N,4] fp32
  float* out = (float*)d_out;                  // [N,7,7,512] fp32
  const int N = in_sizes[1] / 4;               // 1024
  roi_pool_kernel<<<dim3(N * HDIM), dim3(128), 0, stream>>>(fm, rois, out);
}